// MaxPool_42090679501100
// MI455X (gfx1250) — compile-verified
//
#include <hip/hip_runtime.h>
#include <cstdint>
#include <cmath>

// N = 50000 nodes, K = 32 neighbors, D = 128 float32 features.
// One wave32 per node: lane l owns the float4 at column 4*l (covers 128 cols).
// Per neighbor, the wave issues one coalesced 512B row read (32 x B128).

typedef float v4f __attribute__((ext_vector_type(4)));

#define K_NEI           32
#define D_FEAT          128
#define WAVES_PER_BLOCK 8
#define BLOCK_THREADS   (WAVES_PER_BLOCK * 32)
#define CHUNK           8   // neighbor rows in flight per wave

__global__ __launch_bounds__(BLOCK_THREADS)
void gather_max_kernel(const float* __restrict__ s_feats,
                       const long long* __restrict__ nbr_idx,
                       float* __restrict__ out,
                       int n_nodes)
{
    const int wave = threadIdx.x >> 5;
    const int lane = threadIdx.x & 31;
    // Force wave-uniform node id into an SGPR so index loads become s_load_b64
    // through the scalar cache (KMcnt path), keeping VMEM free for the gather.
    const int node = __builtin_amdgcn_readfirstlane(
        (int)(blockIdx.x * WAVES_PER_BLOCK + wave));
    if (node >= n_nodes) return;

    const long long* __restrict__ idx = nbr_idx + (size_t)node * K_NEI;
    const int col = lane << 2;                 // 4 floats per lane

    v4f acc = { -INFINITY, -INFINITY, -INFINITY, -INFINITY };

    for (int c = 0; c < K_NEI; c += CHUNK) {
        // Prefetch next chunk's rows into the cache hierarchy
        // (global_prefetch_b8) while we still owe loads/fmax on this chunk.
        if (c + CHUNK < K_NEI) {
            #pragma unroll
            for (int j = 0; j < CHUNK; ++j) {
                const long long nid = idx[c + CHUNK + j];
                __builtin_prefetch(s_feats + (size_t)nid * D_FEAT + col, 0, 3);
            }
        }

        // Issue all CHUNK coalesced B128 gathers before consuming any:
        // 8 outstanding global_load_b128 per wave for latency hiding.
        v4f v[CHUNK];
        #pragma unroll
        for (int j = 0; j < CHUNK; ++j) {
            const long long nid = idx[c + j];
            v[j] = *(const v4f*)(s_feats + (size_t)nid * D_FEAT + col);
        }

        #pragma unroll
        for (int j = 0; j < CHUNK; ++j) {
            acc.x = fmaxf(acc.x, v[j].x);
            acc.y = fmaxf(acc.y, v[j].y);
            acc.z = fmaxf(acc.z, v[j].z);
            acc.w = fmaxf(acc.w, v[j].w);
        }
    }

    // Streaming output (read-never-again): non-temporal store so the 25.6 MB
    // result doesn't evict the L2-resident feature table.
    __builtin_nontemporal_store(acc, (v4f*)(out + (size_t)node * D_FEAT + col));
}

extern "C" void kernel_launch(void* const* d_in, const int* in_sizes, int n_in,
                              void* d_out, int out_size, void* d_ws, size_t ws_size,
                              hipStream_t stream)
{
    const float*     s_feats = (const float*)d_in[0];      // [N, 128] f32
    const long long* nbr_idx = (const long long*)d_in[1];  // [N, 32]  i64
    float*           out     = (float*)d_out;              // [N, 128] f32

    const int n_nodes = in_sizes[1] / K_NEI;
    const int blocks  = (n_nodes + WAVES_PER_BLOCK - 1) / WAVES_PER_BLOCK;

    gather_max_kernel<<<blocks, BLOCK_THREADS, 0, stream>>>(
        s_feats, nbr_idx, out, n_nodes);
}